// CombinedLoss_41764261986940
// MI455X (gfx1250) — compile-verified
//
#include <hip/hip_runtime.h>
#include <hip/hip_bf16.h>

// Problem constants (baked, match reference)
#define BATCH      32
#define SEQ        1024
#define BIGVAL     1e10f
// exp(x/gamma) = exp2(x * (1/gamma) * log2(e)) ; 10 * 1.4426950408889634
#define EXP_SCALE  14.426950408889634f
// gamma * ln(2) : softmin = m - gamma*ln(s) = m - (gamma*ln2)*log2(s)
#define LOG_SCALE  0.06931471805599453f
// KQTSE * log2(e) = 0.1 * 1.4426950408889634
#define QTSE_EXP_SCALE 0.14426950408889634f

#define QTSE_SCALE (1.0f / 32768.0f)   // 1/(B*T) mean of qtse elements
#define SDTW_SCALE (0.1f / 32.0f)      // SOFT_DTW_WEIGHT / B

typedef __attribute__((ext_vector_type(2))) float        v2f;
typedef __attribute__((ext_vector_type(8))) float        v8f;
typedef __attribute__((ext_vector_type(4))) unsigned int v4u;
typedef __attribute__((ext_vector_type(4))) int          v4i;
typedef __attribute__((ext_vector_type(8))) int          v8i;

#if __has_builtin(__builtin_amdgcn_tensor_load_to_lds)
#define HAVE_TDM 1
#else
#define HAVE_TDM 0
#endif

// ---------------------------------------------------------------------------
// Kernel 1: QTSE partial sums. One block per batch row, 256 threads.
// ws_qtse[b] = sum_k e^2 * exp(0.1 e) over that row.
// ---------------------------------------------------------------------------
__global__ __launch_bounds__(256) void qtse_partial_kernel(
    const float* __restrict__ P, const float* __restrict__ T,
    float* __restrict__ ws_qtse) {
  __shared__ float red[256];
  const int b = blockIdx.x;
  const int t = threadIdx.x;
  const int base = b * SEQ;

  __builtin_prefetch(P + base, 0, 0);   // global_prefetch_b8
  __builtin_prefetch(T + base, 0, 0);

  float acc = 0.0f;
#pragma unroll
  for (int k = t; k < SEQ; k += 256) {
    float e = P[base + k] - T[base + k];
    acc += e * e * __builtin_amdgcn_exp2f(e * QTSE_EXP_SCALE);
  }
  red[t] = acc;
  __syncthreads();
#pragma unroll
  for (int off = 128; off > 0; off >>= 1) {
    if (t < off) red[t] += red[t + off];
    __syncthreads();
  }
  if (t == 0) ws_qtse[b] = red[0];
}

// ---------------------------------------------------------------------------
// Kernel 2: Soft-DTW via anti-diagonal wavefront.
// One block (32 wave32 waves, 1024 threads) per batch element.
// Diagonal d storage Rd[j] = R[d-j, j], three rotating LDS buffers.
// The p row is staged into LDS by the Tensor Data Mover (one DMA per block,
// issued by wave 0, completion via s_wait_tensorcnt); lanes then confirm
// p_sh with an idempotent authoritative write so correctness never depends
// on D# bitfield details.
// ---------------------------------------------------------------------------
__global__ __launch_bounds__(1024) void sdtw_kernel(
    const float* __restrict__ P, const float* __restrict__ T,
    float* __restrict__ ws_sdtw) {
  __shared__ float p_sh[SEQ];
  __shared__ float Rbuf[3][SEQ + 1];   // stride 1025 floats -> conflict-free

  const int b   = blockIdx.x;
  const int tid = threadIdx.x;         // 0..1023
  const int j   = tid + 1;             // this thread owns column j in 1..1024

  __builtin_prefetch(P + b * SEQ, 0, 0);
  __builtin_prefetch(T + b * SEQ, 0, 0);

#if HAVE_TDM
  if (tid < 32) {                      // wave 0 only: TDM is a wave-level DMA
    unsigned int lds_addr = (unsigned int)(size_t)(&p_sh[0]);
    unsigned long long ga = (unsigned long long)(size_t)(P + b * SEQ);
    // D# group 0: count=1 (valid), lds_addr, global_addr[56:0], type=2
    v4u g0;
    g0.x = 1u;
    g0.y = lds_addr;
    g0.z = (unsigned int)(ga & 0xFFFFFFFFu);
    g0.w = (unsigned int)((ga >> 32) & 0x01FFFFFFu) | (2u << 30);
    // D# group 1: mask=0, data_size=2 (4B), 2D tensor 1024x1, tile 1024x1,
    // tensor_dim0_stride=1024, no padding / iterate / atomic-barrier.
    v8i g1;
    g1[0] = (int)(2u << 16);           // data_size=4B
    g1[1] = (int)(1024u << 16);        // tensor_dim0[15:0] in [31:16]
    g1[2] = (int)(1u << 16);           // tensor_dim0[31:16]=0 | tensor_dim1[15:0]=1
    g1[3] = (int)(1024u << 16);        // tensor_dim1[31:16]=0 | tile_dim0=1024
    g1[4] = 1;                         // tile_dim1=1, tile_dim2=0
    g1[5] = 1024;                      // tensor_dim0_stride[31:0]
    g1[6] = 0;                         // stride0[47:32]=0 | stride1[15:0]=0
    g1[7] = 0;
    v4i z4 = {0, 0, 0, 0};
#if __clang_major__ >= 23
    v8i z8 = {0, 0, 0, 0, 0, 0, 0, 0};
    __builtin_amdgcn_tensor_load_to_lds(g0, g1, z4, z4, z8, 0);
#else
    __builtin_amdgcn_tensor_load_to_lds(g0, g1, z4, z4, 0);
#endif
    __builtin_amdgcn_s_wait_tensorcnt(0);
  }
  __syncthreads();                     // TDM data (if any) landed in LDS
#endif

  // Authoritative p_sh contents (idempotent when the TDM wrote the same).
  p_sh[tid] = P[b * SEQ + tid];
  const float tval = T[b * SEQ + tid]; // t_{j-1}, stays in a register

  if (tid == 0) {
    Rbuf[0][0] = 0.0f;     // R[0,0]
    Rbuf[1][0] = BIGVAL;   // R[1,0]
    Rbuf[1][1] = BIGVAL;   // R[0,1]
  }
  __syncthreads();

  int c2 = 0, c1 = 1, c0 = 2;          // prev2, prev, cur buffers
  float result = 0.0f;

  for (int d = 2; d <= 2 * SEQ; ++d) {
    const int jlo = (d - SEQ > 1) ? (d - SEQ) : 1;
    const int jhi = (d - 1 < SEQ) ? (d - 1) : SEQ;
    if (j >= jlo && j <= jhi) {
      const int i = d - j;
      const float diff = p_sh[i - 1] - tval;
      const float Dij  = diff * diff;
      const float a  = Rbuf[c2][j - 1];  // R[i-1, j-1]
      const float bb = Rbuf[c1][j];      // R[i-1, j]
      const float cc = Rbuf[c1][j - 1];  // R[i,   j-1]
      float m = fminf(a, fminf(bb, cc));
      // stable softmin: m - gamma * ln( sum exp(-(x-m)/gamma) )
      float s = __builtin_amdgcn_exp2f((m - a)  * EXP_SCALE)
              + __builtin_amdgcn_exp2f((m - bb) * EXP_SCALE)
              + __builtin_amdgcn_exp2f((m - cc) * EXP_SCALE);
      float r = Dij + m - LOG_SCALE * __builtin_amdgcn_logf(s);
      Rbuf[c0][j] = r;
      if (d == 2 * SEQ) result = r;      // only j==SEQ reaches here: R[N,N]
    }
    // boundary entries of the new diagonal
    if (tid == 0) Rbuf[c0][0] = BIGVAL;             // R[d, 0]
    if (d <= SEQ && tid == 1) Rbuf[c0][d] = BIGVAL; // R[0, d]
    __syncthreads();
    int tmp = c2; c2 = c1; c1 = c0; c0 = tmp;       // rotate
  }

  if (j == SEQ) ws_sdtw[b] = result;
}

// ---------------------------------------------------------------------------
// Kernel 3: final combine. Single wave32; the 64 pre-scaled partials are
// reduced through v_wmma_f32_16x16x4_f32 (A = data, B = ones => D holds
// row sums of A; lanes 0 and 16 together hold all 16 row sums).
// ---------------------------------------------------------------------------
__global__ __launch_bounds__(32) void final_kernel(
    const float* __restrict__ ws, float* __restrict__ out) {
  const int lane = threadIdx.x;  // 0..31, full wave -> EXEC all ones
  const int k0 = 2 * lane, k1 = 2 * lane + 1;
  v2f a;
  a.x = ws[k0] * (k0 < BATCH ? QTSE_SCALE : SDTW_SCALE);
  a.y = ws[k1] * (k1 < BATCH ? QTSE_SCALE : SDTW_SCALE);
  v2f bones; bones.x = 1.0f; bones.y = 1.0f;
  v8f c = {};
  // D = A x B + 0 : emits v_wmma_f32_16x16x4_f32
  v8f d = __builtin_amdgcn_wmma_f32_16x16x4_f32(
      /*neg_a=*/false, a, /*neg_b=*/false, bones,
      /*c_mod=*/(short)0, c, /*reuse_a=*/false, /*reuse_b=*/false);
  float s = 0.0f;
#pragma unroll
  for (int r = 0; r < 8; ++r) s += d[r];
  // lane 0 holds row-sums M=0..7 (column N=0); lane 16 holds M=8..15
  float s_hi = __shfl(s, 16, 32);
  if (lane == 0) out[0] = s + s_hi;
}

// ---------------------------------------------------------------------------
extern "C" void kernel_launch(void* const* d_in, const int* in_sizes, int n_in,
                              void* d_out, int out_size, void* d_ws, size_t ws_size,
                              hipStream_t stream) {
  const float* P = (const float*)d_in[0];   // predicted [32,1024] f32
  const float* T = (const float*)d_in[1];   // target    [32,1024] f32
  float* ws  = (float*)d_ws;                // ws[0..31] qtse, ws[32..63] sdtw
  float* out = (float*)d_out;               // scalar f32

  qtse_partial_kernel<<<BATCH, 256, 0, stream>>>(P, T, ws);
  sdtw_kernel<<<BATCH, SEQ, 0, stream>>>(P, T, ws + BATCH);
  final_kernel<<<1, 32, 0, stream>>>(ws, out);
}